// AddressingHead_26431228739734
// MI455X (gfx1250) — compile-verified
//
#include <hip/hip_runtime.h>
#include <math.h>

// ---------------------------------------------------------------------------
// AddressingHead fused kernel for MI455X (gfx1250, wave32, WMMA)
//   B=256, N=8192, Din=512, MEMORY_BANK_SIZE=128, NUM_SHIFTS=5
// Bandwidth-bound: must stream 1 GiB of `memory` once (~46us @ 23.3 TB/s).
// One block per batch row; everything else lives in LDS (~37 KB/block).
// ---------------------------------------------------------------------------

typedef __attribute__((ext_vector_type(2))) float v2f;
typedef __attribute__((ext_vector_type(8))) float v8f;

#define BATCH   256
#define NMEM    8192
#define DIN     512
#define MBK     128      // MEMORY_BANK_SIZE
#define OUTD    134      // MBK + 1 + NUM_SHIFTS
#define NTHREADS 512     // 16 waves of 32

__global__ __launch_bounds__(NTHREADS)
void addressing_head_fused(const float* __restrict__ input,
                           const float* __restrict__ memory,
                           const float* __restrict__ prev,
                           const float* __restrict__ W,
                           const float* __restrict__ bias,
                           float* __restrict__ out)
{
    __shared__ float sscore[NMEM];                 // scores -> exp -> interp (in place)
    __shared__ __align__(16) float sq[MBK];        // query row
    __shared__ float sin_[DIN];                    // input row
    __shared__ float red[NTHREADS];                // reduction scratch
    __shared__ float sextra[8];                    // args[128..133]
    __shared__ float sgate;
    __shared__ float ssd[5];                       // shift distribution

    const int b = blockIdx.x;
    const int t = threadIdx.x;

    // ---------------- Phase 0: args = input @ W^T + bias (tiny) ------------
    sin_[t] = input[(size_t)b * DIN + t];
    __syncthreads();

    if (t < OUTD) {
        float acc = bias[t];
        const float* wr = W + (size_t)t * DIN;
        #pragma unroll 8
        for (int k = 0; k < DIN; ++k) acc = fmaf(sin_[k], wr[k], acc);
        if (t < MBK) sq[t] = acc;
        else         sextra[t - MBK] = acc;
    }
    __syncthreads();

    if (t == 0) {
        sgate = 1.0f / (1.0f + __expf(-sextra[0]));
        float mx = sextra[1];
        for (int s = 2; s < 6; ++s) mx = fmaxf(mx, sextra[s]);
        float e[5], ssum = 0.f;
        for (int s = 0; s < 5; ++s) { e[s] = __expf(sextra[1 + s] - mx); ssum += e[s]; }
        const float r = 1.0f / ssum;
        for (int s = 0; s < 5; ++s) ssd[s] = e[s] * r;
    }
    __syncthreads();

    // ---------------- Phase 1: scores via V_WMMA_F32_16X16X4_F32 -----------
    // A(16x4) f32 layout: lanes 0-15: M=lane, VGPR0/1 = K0/K1;
    //                     lanes 16-31: M=lane-16, VGPR0/1 = K2/K3.
    // B(4x16): analogous by K-halves; we replicate query across all 16 cols,
    // so every column of D holds the same 16 scores; lanes 0/16 extract D[:,0].
    const int wave = t >> 5;
    const int lane = t & 31;
    const int m16  = lane & 15;
    const int kh   = lane >> 4;                    // 0: K0/K1 half, 1: K2/K3 half
    const float rscale = 0.08838834764831845f;     // 1/sqrt(128)

    for (int tile = wave; tile < NMEM / 16; tile += NTHREADS / 32) {
        const int n0 = tile * 16;
        const float* rowp = memory + (((size_t)b * NMEM + n0 + m16) * MBK) + kh * 2;
        // hint: this wave's next tile is 256 rows ahead
        __builtin_prefetch(rowp + (size_t)256 * MBK, 0, 1);

        v8f c = {0.f, 0.f, 0.f, 0.f, 0.f, 0.f, 0.f, 0.f};
        #pragma unroll 8
        for (int kc = 0; kc < MBK / 4; ++kc) {
            v2f a  = *(const v2f*)(rowp + 4 * kc);           // global_load_b64
            v2f bb = *(const v2f*)(sq + 4 * kc + 2 * kh);    // ds broadcast load
            c = __builtin_amdgcn_wmma_f32_16x16x4_f32(
                    /*neg_a=*/false, a, /*neg_b=*/false, bb,
                    /*c_mod=*/(short)0, c, /*reuse_a=*/false, /*reuse_b=*/false);
        }
        if (m16 == 0) {                                      // lanes 0 and 16
            const int base = n0 + kh * 8;                    // M=0..7 / M=8..15
            #pragma unroll
            for (int v = 0; v < 8; ++v) sscore[base + v] = c[v] * rscale;
        }
    }
    __syncthreads();

    // ---------------- Phase 2: softmax over N, interp, shift ---------------
    float lmax = -3.4e38f;
    for (int n = t; n < NMEM; n += NTHREADS) lmax = fmaxf(lmax, sscore[n]);
    red[t] = lmax;
    __syncthreads();
    for (int off = NTHREADS / 2; off > 0; off >>= 1) {
        if (t < off) red[t] = fmaxf(red[t], red[t + off]);
        __syncthreads();
    }
    const float rowmax = red[0];
    __syncthreads();

    float lsum = 0.f;
    for (int n = t; n < NMEM; n += NTHREADS) {
        float e = __expf(sscore[n] - rowmax);
        sscore[n] = e;
        lsum += e;
    }
    red[t] = lsum;
    __syncthreads();
    for (int off = NTHREADS / 2; off > 0; off >>= 1) {
        if (t < off) red[t] += red[t + off];
        __syncthreads();
    }
    const float inv = 1.0f / red[0];
    const float g = sgate;
    const float* prevrow = prev + (size_t)b * NMEM;
    for (int n = t; n < NMEM; n += NTHREADS) {
        float interp = g * (sscore[n] * inv) + (1.0f - g) * prevrow[n];
        sscore[n] = interp;                                  // same-index RMW: safe
    }
    __syncthreads();

    const float sd0 = ssd[0], sd1 = ssd[1], sd2 = ssd[2], sd3 = ssd[3], sd4 = ssd[4];
    float* outrow = out + (size_t)b * NMEM;
    for (int n = t; n < NMEM; n += NTHREADS) {
        float acc = sd0 * sscore[(n + NMEM - 2) & (NMEM - 1)]
                  + sd1 * sscore[(n + NMEM - 1) & (NMEM - 1)]
                  + sd2 * sscore[n]
                  + sd3 * sscore[(n + 1) & (NMEM - 1)]
                  + sd4 * sscore[(n + 2) & (NMEM - 1)];
        outrow[n] = acc;
    }
}

extern "C" void kernel_launch(void* const* d_in, const int* in_sizes, int n_in,
                              void* d_out, int out_size, void* d_ws, size_t ws_size,
                              hipStream_t stream) {
    (void)n_in; (void)out_size; (void)d_ws; (void)ws_size;
    const float* input  = (const float*)d_in[0];   // (B, 512)
    const float* memory = (const float*)d_in[1];   // (B, 8192, 128)
    const float* prev   = (const float*)d_in[2];   // (B, 8192)
    const float* W      = (const float*)d_in[3];   // (134, 512)
    const float* bias   = (const float*)d_in[4];   // (134,)
    float* out = (float*)d_out;                    // (B, 8192)

    const int B = in_sizes[2] / NMEM;              // 256
    hipLaunchKernelGGL(addressing_head_fused, dim3(B), dim3(NTHREADS), 0, stream,
                       input, memory, prev, W, bias, out);
}